// NonLinearDiffusion_80616536146128
// MI455X (gfx1250) — compile-verified
//
#include <hip/hip_runtime.h>

// Perona-Malik diffusion, 3 iterations fused in one kernel (gfx1250).
// 64x64 output tile per 256-thread block; halo: 8 cols (16B-aligned loads), 6 rows.
// LDS planes u/flux_x/flux_y: 76 rows x 84 pitch fp32 = ~75 KB total.
// Data path: async global->LDS b128 loads (ASYNCcnt) + b128 output stores.

#define W     1024
#define H     1024
#define TILE  64
#define HX    8                    // left/right halo (keeps loads 16B-aligned)
#define HY    6                    // top/bottom halo
#define LWX   (TILE + 2 * HX)      // 80 cols
#define LWY   (TILE + 2 * HY)      // 76 rows
#define LP    (LWX + 4)            // 84: LDS pitch (multiple of 4 words)
#define NQ    (LWX / 4)            // 20 float4 loads per row
#define NT    256

#define KAPPA2 0.0025f             // (0.05)^2
#define DTs    0.25f
#define EPSs   1e-8f

#if defined(__HIP_DEVICE_COMPILE__) && defined(__gfx1250__) && \
    __has_builtin(__builtin_amdgcn_global_load_async_to_lds_b32)
#define HAVE_ASYNC 1
#else
#define HAVE_ASYNC 0
#endif

#if HAVE_ASYNC && __has_builtin(__builtin_amdgcn_global_load_async_to_lds_b128)
#define HAVE_ASYNC128 1
#else
#define HAVE_ASYNC128 0
#endif

#if HAVE_ASYNC
typedef __attribute__((address_space(1))) int* gasptr_t;
typedef __attribute__((address_space(3))) int* lasptr_t;
#endif
#if HAVE_ASYNC128
typedef int v4i_t __attribute__((__vector_size__(4 * sizeof(int))));
typedef __attribute__((address_space(1))) v4i_t* gasptr4_t;
typedef __attribute__((address_space(3))) v4i_t* lasptr4_t;
#endif

__device__ __forceinline__ void wait_async_zero() {
#if HAVE_ASYNC
#if __has_builtin(__builtin_amdgcn_s_wait_asynccnt)
  __builtin_amdgcn_s_wait_asynccnt(0);
#else
  asm volatile("s_wait_asynccnt 0" ::: "memory");
#endif
#endif
}

__global__ __launch_bounds__(NT) void pm_fused3(const float* __restrict__ img,
                                                float* __restrict__ out) {
  __shared__ __align__(16) float su [LWY * LP];
  __shared__ __align__(16) float sfx[LWY * LP];
  __shared__ __align__(16) float sfy[LWY * LP];

  const int tid = threadIdx.x;
  const int tx0 = (int)blockIdx.x * TILE - HX;  // global x of LDS col 0 (mult of 4)
  const int ty0 = (int)blockIdx.y * TILE - HY;  // global y of LDS row 0
  const long long plane = (long long)blockIdx.z * (long long)(W * H);
  const float* __restrict__ gin  = img + plane;
  float*       __restrict__ gout = out + plane;

  const bool interior =
      (tx0 >= 0) && (ty0 >= 0) && (tx0 + LWX <= W) && (ty0 + LWY <= H);

  // ---- Zero-fill only for border tiles (emulates SAME zero padding).
  if (!interior) {
    for (int i = tid; i < LWY * LP; i += NT) su[i] = 0.0f;
    __syncthreads();
  }

  // ---- Load 76x80 tile as aligned float4s via async global->LDS.
  // Alignment (tx0 % 4 == 0, W % 4 == 0) guarantees a float4 never straddles
  // the image edge, so the per-quad bounds check is exact.
  for (int i = tid; i < LWY * NQ; i += NT) {
    const int r  = i / NQ;
    const int q  = i - r * NQ;
    const int gy = ty0 + r;
    const int gx = tx0 + 4 * q;
    if (((unsigned)gy < (unsigned)H) & ((unsigned)gx < (unsigned)W)) {
      const float* gsrc = gin + (size_t)gy * W + gx;
      float* ldst = &su[r * LP + 4 * q];
#if HAVE_ASYNC128
      __builtin_amdgcn_global_load_async_to_lds_b128((gasptr4_t)gsrc,
                                                     (lasptr4_t)ldst, 0, 0);
#elif HAVE_ASYNC
      __builtin_amdgcn_global_load_async_to_lds_b32((gasptr_t)(gsrc + 0),
                                                    (lasptr_t)(ldst + 0), 0, 0);
      __builtin_amdgcn_global_load_async_to_lds_b32((gasptr_t)(gsrc + 1),
                                                    (lasptr_t)(ldst + 1), 0, 0);
      __builtin_amdgcn_global_load_async_to_lds_b32((gasptr_t)(gsrc + 2),
                                                    (lasptr_t)(ldst + 2), 0, 0);
      __builtin_amdgcn_global_load_async_to_lds_b32((gasptr_t)(gsrc + 3),
                                                    (lasptr_t)(ldst + 3), 0, 0);
#else
      *(float4*)ldst = *(const float4*)gsrc;
#endif
    }
  }
  wait_async_zero();
  __syncthreads();

  // Divergence of (flux_x, flux_y), reference Sobel taps (cross-correlation).
  auto divergence = [&](int k) -> float {
    const float dvx = ((sfx[k - LP + 1] - sfx[k - LP - 1]) +
                       2.0f * (sfx[k + 1] - sfx[k - 1]) +
                       (sfx[k + LP + 1] - sfx[k + LP - 1])) * 0.125f;
    const float dvy = ((sfy[k + LP - 1] - sfy[k - LP - 1]) +
                       2.0f * (sfy[k + LP] - sfy[k - LP]) +
                       (sfy[k + LP + 1] - sfy[k - LP + 1])) * 0.125f;
    return dvx + dvy;
  };

#pragma unroll
  for (int it = 0; it < 3; ++it) {
    // Stage A: flux over x in [a0x, a0x+RA), y in [a0y, a0y+RA).
    const int a0x = 3 + 2 * it;      // x-halo is 8, output cols start at 8
    const int a0y = 1 + 2 * it;      // y-halo is 6, output rows start at 6
    const int RA  = 74 - 4 * it;     // 74, 70, 66 (constant after unroll)
    for (int i = tid; i < RA * RA; i += NT) {
      const int yy = i / RA;
      const int y  = a0y + yy;
      const int x  = a0x + (i - yy * RA);
      const int k  = y * LP + x;
      const float tL = su[k - LP - 1], tC = su[k - LP], tR = su[k - LP + 1];
      const float mL = su[k - 1],                       mR = su[k + 1];
      const float bL = su[k + LP - 1], bC = su[k + LP], bR = su[k + LP + 1];
      const float gxv = ((tR - tL) + 2.0f * (mR - mL) + (bR - bL)) * 0.125f;
      const float gyv = ((bL - tL) + 2.0f * (bC - tC) + (bR - tR)) * 0.125f;
      const float s   = gxv * gxv + gyv * gyv + EPSs;
      float cnd = KAPPA2 / (KAPPA2 + s);  // == 1/(1+(|g|/kappa)^2), eps inside
      if (!interior) {
        // Reference zero-pads flux for the second conv: out-of-image flux = 0.
        const bool inimg = ((unsigned)(ty0 + y) < (unsigned)H) &
                           ((unsigned)(tx0 + x) < (unsigned)W);
        cnd = inimg ? cnd : 0.0f;
      }
      sfx[k] = cnd * gxv;
      sfy[k] = cnd * gyv;
    }
    __syncthreads();

    // Stage B: u += DT * div(flux), one pixel inset from Stage A region.
    const int b0x = a0x + 1;
    const int b0y = a0y + 1;
    const int RB  = RA - 2;          // 72, 68, 64
    if (it < 2) {
      for (int i = tid; i < RB * RB; i += NT) {
        const int yy = i / RB;
        const int y  = b0y + yy;
        const int x  = b0x + (i - yy * RB);
        const int k  = y * LP + x;
        const float unew = su[k] + DTs * divergence(k);
        if (interior) {
          su[k] = unew;
        } else {
          // Out-of-image halo pixels must stay exactly zero for later iters.
          const bool inimg = ((unsigned)(ty0 + y) < (unsigned)H) &
                             ((unsigned)(tx0 + x) < (unsigned)W);
          if (inimg) su[k] = unew;
        }
      }
      __syncthreads();
    } else {
      // Final iteration: B-region == output tile (x in [8,72), y in [6,70)).
      // Output rows start at 64*bx (16B-aligned) -> vector b128 stores.
      const int NQ4 = TILE / 4;  // 16 float4s per output row
      for (int i = tid; i < TILE * NQ4; i += NT) {
        const int r = i / NQ4;
        const int q = i - r * NQ4;
        const int y = HY + r;
        const int x = HX + 4 * q;
        const int k = y * LP + x;
        float4 v;
        v.x = su[k + 0] + DTs * divergence(k + 0);
        v.y = su[k + 1] + DTs * divergence(k + 1);
        v.z = su[k + 2] + DTs * divergence(k + 2);
        v.w = su[k + 3] + DTs * divergence(k + 3);
        *(float4*)(gout + (size_t)(ty0 + y) * W + (tx0 + x)) = v;
      }
    }
  }
}

extern "C" void kernel_launch(void* const* d_in, const int* in_sizes, int n_in,
                              void* d_out, int out_size, void* d_ws,
                              size_t ws_size, hipStream_t stream) {
  (void)n_in; (void)out_size; (void)d_ws; (void)ws_size;
  const float* img = (const float*)d_in[0];
  float* out = (float*)d_out;
  const int nplanes = in_sizes[0] / (W * H);  // 16
  dim3 grid(W / TILE, H / TILE, nplanes);
  pm_fused3<<<grid, dim3(NT, 1, 1), 0, stream>>>(img, out);
}